// FeedForwarder_32942399160740
// MI455X (gfx1250) — compile-verified
//
#include <hip/hip_runtime.h>

// FFN (x@W1+b1)@W2+b2 + x, then LayerNorm over E. Per-position weights.
// B=32, S=128, E=512, F=2048. HBM-bound (~1.07GB weights, 17.2 GFLOP,
// 16 flop/byte << fp32-WMMA ridge) -> stream weights once, fp32 WMMA.
#define B_ 32
#define S_ 128
#define E_ 512
#define F_ 2048
#define EPS_ 1e-5f

typedef __attribute__((ext_vector_type(2))) float v2f;
typedef __attribute__((ext_vector_type(8))) float v8f;

// XOR swizzle (flips addr bits [5:2]) so 16-lane column gathers with row
// stride 512/2048 floats hit 16 distinct LDS banks.
__device__ __forceinline__ int swz4(int m) { return (m & 15) << 2; }

__launch_bounds__(1024)
__global__ void ffn_ln_kernel(const float* __restrict__ x,
                              const float* __restrict__ W1,
                              const float* __restrict__ b1,
                              const float* __restrict__ W2,
                              const float* __restrict__ b2,
                              const float* __restrict__ gamma,
                              const float* __restrict__ beta,
                              float* __restrict__ out) {
  extern __shared__ float lds[];          // 320 KB total (WGP max)
  float* ldsX = lds;                      // [32][512]  = 64 KB
  float* ldsH = lds + B_ * E_;            // [32][2048] = 256 KB

  const int tid  = threadIdx.x;
  const int lane = tid & 31;
  const int wave = tid >> 5;              // 0..31
  const int s    = blockIdx.x;            // 0..127

  const float* W1s = W1 + (size_t)s * E_ * F_;
  const float* W2s = W2 + (size_t)s * F_ * E_;
  const float* b1s = b1 + (size_t)s * F_;
  const float* b2s = b2 + (size_t)s * E_;

  // ---- Phase 0: stage X[:, s, :] (32x512) into LDS, swizzled ----
  for (int i = tid; i < B_ * E_; i += 1024) {
    const int m = i >> 9;                 // row (batch)
    const int e = i & (E_ - 1);
    ldsX[m * E_ + (e ^ swz4(m))] = x[((size_t)m * S_ + s) * E_ + e];
  }
  __syncthreads();

  const int lh = lane & 15;               // position within half-wave
  const int hi = lane >> 4;               // half-wave select

  // ---- Phase 1: H = X @ W1_s + b1_s ; wave owns 64 F-columns ----
  {
    v8f acc[2][4] = {};                   // 2 M-tiles x 4 N-tiles
    const int fbase0 = wave * 64;
    for (int kk = 0; kk < E_ / 4; ++kk) {
      // A operand 16x4 f32: VGPR0/1 = K k0/k0+1; upper half-wave K+2.
      const int k0 = kk * 4 + hi * 2;
      v2f a[2];
#pragma unroll
      for (int mt = 0; mt < 2; ++mt) {
        const int m = mt * 16 + lh;
        a[mt] = *(const v2f*)&ldsX[m * E_ + (k0 ^ swz4(m))];
      }
#pragma unroll
      for (int nt = 0; nt < 4; ++nt) {
        // B operand 4x16 f32: VGPR r -> K row r (upper half-wave +2), N=lane.
        const int f = fbase0 + nt * 16 + lh;
        v2f b;
        b.x = W1s[(size_t)(kk * 4 + hi * 2 + 0) * F_ + f];
        b.y = W1s[(size_t)(kk * 4 + hi * 2 + 1) * F_ + f];
#pragma unroll
        for (int mt = 0; mt < 2; ++mt)
          acc[mt][nt] = __builtin_amdgcn_wmma_f32_16x16x4_f32(
              false, a[mt], false, b, (short)0, acc[mt][nt], false, false);
      }
    }
    // Epilogue: +b1, store H tiles to LDS (swizzled).
    // C layout: VGPR r -> row M=r (upper half-wave M=r+8), N=lane.
#pragma unroll
    for (int nt = 0; nt < 4; ++nt) {
      const int f = fbase0 + nt * 16 + lh;
      const float bias = b1s[f];
#pragma unroll
      for (int mt = 0; mt < 2; ++mt)
#pragma unroll
        for (int r = 0; r < 8; ++r) {
          const int m = mt * 16 + r + hi * 8;
          ldsH[m * F_ + (f ^ swz4(m))] = acc[mt][nt][r] + bias;
        }
    }
  }
  __syncthreads();

  // ---- Phase 2: Y = H @ W2_s + b2_s + X ; wave owns 16 E-columns ----
  v8f yacc[2] = {};
  const int ebase = wave * 16;
  for (int kk = 0; kk < F_ / 4; ++kk) {
    const int k0 = kk * 4 + hi * 2;
    v2f a[2];
#pragma unroll
    for (int mt = 0; mt < 2; ++mt) {
      const int m = mt * 16 + lh;
      a[mt] = *(const v2f*)&ldsH[m * F_ + (k0 ^ swz4(m))];
    }
    const int e = ebase + lh;
    v2f b;
    b.x = W2s[(size_t)(kk * 4 + hi * 2 + 0) * E_ + e];
    b.y = W2s[(size_t)(kk * 4 + hi * 2 + 1) * E_ + e];
#pragma unroll
    for (int mt = 0; mt < 2; ++mt)
      yacc[mt] = __builtin_amdgcn_wmma_f32_16x16x4_f32(
          false, a[mt], false, b, (short)0, yacc[mt], false, false);
  }
  {
    const int e = ebase + lh;
    const float bias2 = b2s[e];
#pragma unroll
    for (int mt = 0; mt < 2; ++mt)
#pragma unroll
      for (int r = 0; r < 8; ++r) {
        const int m = mt * 16 + r + hi * 8;
        yacc[mt][r] += bias2 + ldsX[m * E_ + (e ^ swz4(m))];
      }
  }
  __syncthreads();  // all waves done reading ldsH -> reuse it for reduction

  // ---- Phase 3: LayerNorm over E=512 per batch row ----
  float* redS  = ldsH;            // [32 rows][32 waves] partial sums
  float* redQ  = ldsH + 1024;     // [32][32] partial sum-of-squares
  float* stats = ldsH + 2048;     // [32][{mean, rstd}]

#pragma unroll
  for (int mt = 0; mt < 2; ++mt)
#pragma unroll
    for (int r = 0; r < 8; ++r) {
      float s1 = yacc[mt][r];
      float s2 = s1 * s1;
#pragma unroll
      for (int mask = 1; mask < 16; mask <<= 1) {   // reduce within half-wave
        s1 += __shfl_xor(s1, mask, 32);
        s2 += __shfl_xor(s2, mask, 32);
      }
      const int m = mt * 16 + r + hi * 8;
      if (lh == 0) {                                // lanes 0 and 16 write
        redS[m * 32 + wave] = s1;
        redQ[m * 32 + wave] = s2;
      }
    }
  __syncthreads();

  {  // wave w reduces row w across the 32 per-wave partials
    float s1 = redS[wave * 32 + lane];
    float s2 = redQ[wave * 32 + lane];
#pragma unroll
    for (int mask = 1; mask < 32; mask <<= 1) {
      s1 += __shfl_xor(s1, mask, 32);
      s2 += __shfl_xor(s2, mask, 32);
    }
    if (lane == 0) {
      const float mean = s1 * (1.0f / E_);
      const float var  = s2 * (1.0f / E_) - mean * mean;
      stats[wave * 2 + 0] = mean;
      stats[wave * 2 + 1] = rsqrtf(var + EPS_);
    }
  }
  __syncthreads();

  {  // normalize, affine, store
    const int e = ebase + lh;
    const float g  = gamma[e];
    const float bb = beta[e];
#pragma unroll
    for (int mt = 0; mt < 2; ++mt)
#pragma unroll
      for (int r = 0; r < 8; ++r) {
        const int m = mt * 16 + r + hi * 8;
        const float mean = stats[m * 2 + 0];
        const float rstd = stats[m * 2 + 1];
        out[((size_t)m * S_ + s) * E_ + e] =
            (yacc[mt][r] - mean) * rstd * g + bb;
      }
  }
}

extern "C" void kernel_launch(void* const* d_in, const int* in_sizes, int n_in,
                              void* d_out, int out_size, void* d_ws, size_t ws_size,
                              hipStream_t stream) {
  const float* x     = (const float*)d_in[0];
  const float* W1    = (const float*)d_in[1];
  const float* b1    = (const float*)d_in[2];
  const float* W2    = (const float*)d_in[3];
  const float* b2    = (const float*)d_in[4];
  const float* gamma = (const float*)d_in[5];
  const float* beta  = (const float*)d_in[6];
  float* out = (float*)d_out;

  const size_t shmem = (size_t)(B_ * E_ + B_ * F_) * sizeof(float);  // 320 KB
  ffn_ln_kernel<<<dim3(S_), dim3(1024), shmem, stream>>>(
      x, W1, b1, W2, b2, gamma, beta, out);
}